// MyEGNNNet_64141041598615
// MI455X (gfx1250) — compile-verified
//
#include <hip/hip_runtime.h>
#include <math.h>

// Problem constants (from reference)
#define N_NODES 50000
#define BATCH   2
#define C_DIM   64
#define NB      (N_NODES * BATCH)   // 100000 rows of length 64
#define E_EDGES 800000

typedef __attribute__((ext_vector_type(2))) float v2f;
typedef __attribute__((ext_vector_type(8))) float v8f;

__device__ __forceinline__ float sigmoidf_fast(float v) {
    return 1.0f / (1.0f + __expf(-v));
}

// Stage nfloats (multiple of 1024) from global to LDS with a 256-thread block.
// Uses the CDNA5 async global->LDS DMA path (ASYNCcnt-tracked) via inline asm;
// each lane issues a 16B async copy.  Ends with a full block barrier.
__device__ __forceinline__ void lds_stage(const float* __restrict__ g,
                                          float* l, int nfloats) {
#if defined(__gfx1250__)
    {
        const int bytes = nfloats * 4;
        // Low 32 bits of a flat shared pointer are the LDS byte offset
        // (ISA 10.2, LDS aperture).
        unsigned lbase = (unsigned)(uintptr_t)l;
        const char* gb = (const char*)g;
        for (int off = (int)threadIdx.x * 16; off < bytes; off += 256 * 16) {
            unsigned lds_addr = lbase + (unsigned)off;
            const void* gaddr = (const void*)(gb + off);
            asm volatile("global_load_async_to_lds_b128 %0, %1, off"
                         :: "v"(lds_addr), "v"(gaddr)
                         : "memory");
        }
        asm volatile("s_wait_asynccnt 0x0" ::: "memory");
    }
#else
    for (int i = threadIdx.x; i < nfloats / 4; i += 256)
        ((float4*)l)[i] = ((const float4*)g)[i];
#endif
    __syncthreads();
}

// ---------------------------------------------------------------------------
// Kernel 0a: tiny prep.  vq[c] = sum_d query_w[c,d]*att_w[d]
//                        vk[c] = sum_d key_w[c,d]*att_w[C+d]
//                        c0    = sum_c weight_e[c]*att_w[2C+c]
// ---------------------------------------------------------------------------
__global__ void prep_kernel(const float* __restrict__ qw,
                            const float* __restrict__ kw,
                            const float* __restrict__ attw,
                            const float* __restrict__ we,
                            float* __restrict__ vq,
                            float* __restrict__ vk,
                            float* __restrict__ c0) {
    int c = threadIdx.x;  // 64 threads
    float sq = 0.f, sk = 0.f;
    #pragma unroll 8
    for (int d = 0; d < C_DIM; ++d) {
        sq += qw[c * C_DIM + d] * attw[d];
        sk += kw[c * C_DIM + d] * attw[C_DIM + d];
    }
    vq[c] = sq;
    vk[c] = sk;
    if (c == 0) {
        float s = 0.f;
        for (int i = 0; i < C_DIM; ++i) s += we[i] * attw[2 * C_DIM + i];
        *c0 = s;
    }
}

// ---------------------------------------------------------------------------
// Kernel 0b: swizzle the three 64x64 weight matrices into WMMA B-fragment
// order so the GEMM inner loops read one contiguous float2 per fragment
// (single ds_load_b64, no register marshalling):
//   frag f = (kk*4 + ct)*32 + lane,  lane = half*16 + l15
//   dst[f] = ( W[kk*4 + half*2    ][ct*16 + l15],
//              W[kk*4 + half*2 + 1][ct*16 + l15] )
// m=0 -> weight_n, m=1 -> out_w[:C], m=2 -> out_w[C:]
// ---------------------------------------------------------------------------
__global__ void swizzle_kernel(const float* __restrict__ Wn,
                               const float* __restrict__ outw,
                               float* __restrict__ Wn_f,
                               float* __restrict__ W12_f) {
    int t = blockIdx.x * blockDim.x + threadIdx.x;  // 0 .. 3*2048-1
    if (t >= 3 * 2048) return;
    const int m    = t >> 11;
    const int f    = t & 2047;
    const int kk   = f >> 7;
    const int ct   = (f >> 5) & 3;
    const int lane = f & 31;
    const int half = lane >> 4;
    const int l15  = lane & 15;
    const int k0   = kk * 4 + half * 2;
    const int col  = ct * 16 + l15;
    const float* src = (m == 0) ? Wn : (outw + (m - 1) * C_DIM * C_DIM);
    float*       dst = (m == 0) ? Wn_f : (W12_f + (m - 1) * C_DIM * C_DIM);
    dst[f * 2 + 0] = src[k0 * C_DIM + col];
    dst[f * 2 + 1] = src[(k0 + 1) * C_DIM + col];
}

// ---------------------------------------------------------------------------
// Kernel 1: x = X @ Wn   via V_WMMA_F32_16X16X4_F32.
// One wave computes a 16-row x 64-col output tile (4 accumulators, 16 K-steps).
// A frag (16x4 f32): lanes 0-15 hold M=lane, K={0,1}; lanes 16-31 M=lane-16,
// K={2,3}.  B frags come pre-swizzled from LDS.  C/D: VGPR r ->
// M = r + 8*(lane>=16), N = lane&15.
// ---------------------------------------------------------------------------
__global__ __launch_bounds__(256) void node_gemm_kernel(
        const float* __restrict__ X, const float* __restrict__ Wn_f,
        float* __restrict__ x) {
    __shared__ float Wl[2048 * 2];  // 16 KB, fragment order
    lds_stage(Wn_f, Wl, 2048 * 2);
    const v2f* Wf = (const v2f*)Wl;

    const int wave    = threadIdx.x >> 5;
    const int lane    = threadIdx.x & 31;
    const int half    = lane >> 4;
    const int l15     = lane & 15;
    const int rowBase = (blockIdx.x * 8 + wave) * 16;

    int arow = rowBase + l15;
    if (arow >= NB) arow = NB - 1;      // clamp (WMMA needs full EXEC)
    const float* Xrow = X + (long)arow * C_DIM;

    v8f acc[4] = {};
    #pragma unroll
    for (int kk = 0; kk < 16; ++kk) {
        const int k0 = kk * 4 + half * 2;
        v2f a;
        a.x = Xrow[k0];
        a.y = Xrow[k0 + 1];
        #pragma unroll
        for (int ct = 0; ct < 4; ++ct) {
            const v2f b = Wf[(kk * 4 + ct) * 32 + lane];
            acc[ct] = __builtin_amdgcn_wmma_f32_16x16x4_f32(
                false, a, false, b, (short)0, acc[ct], false, false);
        }
    }

    #pragma unroll
    for (int r = 0; r < 8; ++r) {
        const int row = rowBase + r + half * 8;
        if (row < NB) {
            #pragma unroll
            for (int ct = 0; ct < 4; ++ct)
                x[(long)row * C_DIM + ct * 16 + l15] = acc[ct][r];
        }
    }
}

// ---------------------------------------------------------------------------
// Kernel 2: per-row attention scalars  aq[row] = x[row]·vq, ak[row] = x[row]·vk
// ---------------------------------------------------------------------------
__global__ __launch_bounds__(256) void dots_kernel(
        const float* __restrict__ x, const float* __restrict__ vq,
        const float* __restrict__ vk, float* __restrict__ aq,
        float* __restrict__ ak) {
    const int row = blockIdx.x * blockDim.x + threadIdx.x;
    if (row >= NB) return;
    const float* xr = x + (long)row * C_DIM;
    float sq = 0.f, sk = 0.f;
    #pragma unroll 8
    for (int c = 0; c < C_DIM; ++c) {
        const float v = xr[c];
        sq += v * vq[c];
        sk += v * vk[c];
    }
    aq[row] = sq;
    ak[row] = sk;
}

// ---------------------------------------------------------------------------
// Kernel 3: edge scatter.  One wave per edge; lanes 0-15 -> b=0, 16-31 -> b=1,
// each lane owns a 4-float chunk of the 64-channel row (coalesced 512B/edge).
// logit = aq[src,b] + ak[dst,b] + ew*c0 + att_b
// aggr[dst,b,:] += sigmoid(logit) * x[src,b,:] * sigmoid(ew * weight_e[:])
// x and aggr are L2-resident (25.6 MB each in 192 MB L2).
// ---------------------------------------------------------------------------
__global__ __launch_bounds__(256) void edge_kernel(
        const int* __restrict__ ei, const float* __restrict__ ew,
        const float* __restrict__ we, const float* __restrict__ x,
        const float* __restrict__ aq, const float* __restrict__ ak,
        const float* __restrict__ c0p, const float* __restrict__ attb,
        float* __restrict__ aggr) {
    const int e = blockIdx.x * 8 + (threadIdx.x >> 5);
    if (e >= E_EDGES) return;
    const int lane  = threadIdx.x & 31;
    const int b     = lane >> 4;
    const int cbase = (lane & 15) * 4;

    const int   src = ei[e];
    const int   dst = ei[E_EDGES + e];
    const float w   = ew[e];

    const float logit = aq[src * BATCH + b] + ak[dst * BATCH + b]
                      + w * c0p[0] + attb[0];
    const float att = sigmoidf_fast(logit);

    const float4 xj  = *reinterpret_cast<const float4*>(
        x + ((long)src * BATCH + b) * C_DIM + cbase);
    const float4 wev = *reinterpret_cast<const float4*>(we + cbase);

    float4 m;
    m.x = att * xj.x * sigmoidf_fast(w * wev.x);
    m.y = att * xj.y * sigmoidf_fast(w * wev.y);
    m.z = att * xj.z * sigmoidf_fast(w * wev.z);
    m.w = att * xj.w * sigmoidf_fast(w * wev.w);

    float* dp = aggr + ((long)dst * BATCH + b) * C_DIM + cbase;
    atomicAdd(dp + 0, m.x);
    atomicAdd(dp + 1, m.y);
    atomicAdd(dp + 2, m.z);
    atomicAdd(dp + 3, m.w);
}

// ---------------------------------------------------------------------------
// Kernel 4: out = x + x@W1 + aggr@W2 + out_b   (W1 = out_w[:C], W2 = out_w[C:])
// Same WMMA tiling as kernel 1, two chained GEMMs into one accumulator.
// ---------------------------------------------------------------------------
__global__ __launch_bounds__(256) void out_gemm_kernel(
        const float* __restrict__ x, const float* __restrict__ aggr,
        const float* __restrict__ W12_f, const float* __restrict__ out_b,
        float* __restrict__ out) {
    __shared__ float Wl[2 * 2048 * 2];  // 32 KB: W1 frags then W2 frags
    lds_stage(W12_f, Wl, 2 * 2048 * 2);
    const v2f* Wf = (const v2f*)Wl;

    const int wave    = threadIdx.x >> 5;
    const int lane    = threadIdx.x & 31;
    const int half    = lane >> 4;
    const int l15     = lane & 15;
    const int rowBase = (blockIdx.x * 8 + wave) * 16;

    int arow = rowBase + l15;
    if (arow >= NB) arow = NB - 1;
    const float* xrow = x    + (long)arow * C_DIM;
    const float* grow = aggr + (long)arow * C_DIM;

    v8f acc[4] = {};
    #pragma unroll
    for (int kk = 0; kk < 16; ++kk) {
        const int k0 = kk * 4 + half * 2;
        v2f aX, aG;
        aX.x = xrow[k0];
        aX.y = xrow[k0 + 1];
        aG.x = grow[k0];
        aG.y = grow[k0 + 1];
        #pragma unroll
        for (int ct = 0; ct < 4; ++ct) {
            const v2f b1 = Wf[(kk * 4 + ct) * 32 + lane];
            const v2f b2 = Wf[2048 + (kk * 4 + ct) * 32 + lane];
            acc[ct] = __builtin_amdgcn_wmma_f32_16x16x4_f32(
                false, aX, false, b1, (short)0, acc[ct], false, false);
            acc[ct] = __builtin_amdgcn_wmma_f32_16x16x4_f32(
                false, aG, false, b2, (short)0, acc[ct], false, false);
        }
    }

    #pragma unroll
    for (int r = 0; r < 8; ++r) {
        const int row = rowBase + r + half * 8;
        if (row < NB) {
            #pragma unroll
            for (int ct = 0; ct < 4; ++ct) {
                const int col = ct * 16 + l15;
                out[(long)row * C_DIM + col] =
                    acc[ct][r] + x[(long)row * C_DIM + col] + out_b[col];
            }
        }
    }
}

// ---------------------------------------------------------------------------
// Host-side launch.  Input order: X, edge_index, edge_weight, weight_n,
// weight_e, query_w, key_w, att_w, att_b, out_w, out_b.
// ---------------------------------------------------------------------------
extern "C" void kernel_launch(void* const* d_in, const int* in_sizes, int n_in,
                              void* d_out, int out_size, void* d_ws,
                              size_t ws_size, hipStream_t stream) {
    const float* X    = (const float*)d_in[0];
    const int*   ei   = (const int*)  d_in[1];   // (2, E) int32
    const float* ew   = (const float*)d_in[2];
    const float* Wn   = (const float*)d_in[3];
    const float* we   = (const float*)d_in[4];
    const float* qw   = (const float*)d_in[5];
    const float* kw   = (const float*)d_in[6];
    const float* attw = (const float*)d_in[7];
    const float* attb = (const float*)d_in[8];
    const float* outw = (const float*)d_in[9];
    const float* outb = (const float*)d_in[10];
    float* out = (float*)d_out;

    // Workspace layout (all fp32, 16B-aligned sections):
    float* x     = (float*)d_ws;                   // NB*64   (25.6 MB)
    float* aggr  = x     + (size_t)NB * C_DIM;     // NB*64   (25.6 MB)
    float* Wn_f  = aggr  + (size_t)NB * C_DIM;     // 4096    (swizzled Wn)
    float* W12_f = Wn_f  + 4096;                   // 8192    (swizzled out_w)
    float* aq    = W12_f + 8192;                   // NB
    float* ak    = aq    + NB;                     // NB
    float* vq    = ak    + NB;                     // 64
    float* vk    = vq    + C_DIM;                  // 64
    float* c0    = vk    + C_DIM;                  // 1

    (void)hipMemsetAsync(aggr, 0, (size_t)NB * C_DIM * sizeof(float), stream);

    prep_kernel<<<1, 64, 0, stream>>>(qw, kw, attw, we, vq, vk, c0);
    swizzle_kernel<<<24, 256, 0, stream>>>(Wn, outw, Wn_f, W12_f);

    const int rowBlocks = (NB + 127) / 128;  // 128 rows (8 waves x 16) / block
    node_gemm_kernel<<<rowBlocks, 256, 0, stream>>>(X, Wn_f, x);

    dots_kernel<<<(NB + 255) / 256, 256, 0, stream>>>(x, vq, vk, aq, ak);

    edge_kernel<<<(E_EDGES + 7) / 8, 256, 0, stream>>>(ei, ew, we, x, aq, ak,
                                                       c0, attb, aggr);

    out_gemm_kernel<<<rowBlocks, 256, 0, stream>>>(x, aggr, W12_f, outb, out);
}